// DownUpLayer_23596550324549
// MI455X (gfx1250) — compile-verified
//
#include <hip/hip_runtime.h>

// Problem constants (match reference).
#define NNODES 50000
#define NEDGES 625000
#define HDIM   128
#define BDIM   32
#define NCNT   (2 * NNODES)   // concatenated down|up bucket counters

typedef float v2f __attribute__((ext_vector_type(2)));
typedef float v8f __attribute__((ext_vector_type(8)));

// D = A(16x4,f32) * B(4x16,f32) + C(16x16,f32), wave32.
// A layout: lane&15 = M, lane>>4 selects K-pair {0,1}|{2,3}; v0=K even, v1=K odd.
// B layout (symmetric): lane&15 = N, lane>>4 selects K-pair; v0=K even, v1=K odd.
// C/D layout: VGPR r -> M = r + 8*(lane>>4), N = lane&15.
static __device__ __forceinline__ v8f wmma_f32(v2f a, v2f b, v8f c) {
  return __builtin_amdgcn_wmma_f32_16x16x4_f32(false, a, false, b, (short)0, c,
                                               false, false);
}

// ---------------------------------------------------------------------------
// Stage A1: degree histogram for both directions (int atomics, 2 per edge).
// cnt[0..N)   = in-degree  (down direction, bucketed by dst)
// cnt[N..2N)  = out-degree (up direction,   bucketed by src)
// ---------------------------------------------------------------------------
__global__ __launch_bounds__(256) void degree_hist(
    const int* __restrict__ ei, int* __restrict__ cnt) {
  const int e = blockIdx.x * blockDim.x + threadIdx.x;
  if (e >= NEDGES) return;
  atomicAdd(&cnt[ei[NEDGES + e]], 1);
  atomicAdd(&cnt[NNODES + ei[e]], 1);
}

// ---------------------------------------------------------------------------
// Stage A2: exclusive prefix scan over the 100k counters. Single workgroup,
// 1024 threads, LDS Hillis-Steele per 1024-chunk with running carry.
// ---------------------------------------------------------------------------
__global__ __launch_bounds__(1024) void scan_exclusive(
    const int* __restrict__ cnt, int* __restrict__ offs) {
  __shared__ int sh[1024];
  const int t = threadIdx.x;
  int carry = 0;
  for (int base = 0; base < NCNT; base += 1024) {
    const int i = base + t;
    const int v = (i < NCNT) ? cnt[i] : 0;
    sh[t] = v;
    __syncthreads();
    int acc = v;
#pragma unroll
    for (int off = 1; off < 1024; off <<= 1) {
      const int add = (t >= off) ? sh[t - off] : 0;
      __syncthreads();
      acc += add;
      sh[t] = acc;
      __syncthreads();
    }
    if (i < NCNT) offs[i] = carry + acc - v;  // exclusive
    carry += sh[1023];
    __syncthreads();
  }
}

// ---------------------------------------------------------------------------
// Stage A3: bucket fill via atomic cursors (cursor[] starts as copy of offs).
// srcbuf[0..E)  : source node of each down-edge, grouped by dst
// srcbuf[E..2E) : dest node of each up-edge, grouped by src
// ---------------------------------------------------------------------------
__global__ __launch_bounds__(256) void fill_buckets(
    const int* __restrict__ ei, int* __restrict__ cursor,
    int* __restrict__ srcbuf) {
  const int e = blockIdx.x * blockDim.x + threadIdx.x;
  if (e >= NEDGES) return;
  const int s = ei[e];
  const int d = ei[NEDGES + e];
  const int p = atomicAdd(&cursor[d], 1);
  srcbuf[p] = s;
  const int q = atomicAdd(&cursor[NNODES + s], 1);
  srcbuf[q] = d;
}

// ---------------------------------------------------------------------------
// Stage A4: per-node gather-sum (replaces 160M fp32 atomics with pure
// register-accumulated float4 reads from L2). One wave per (node, direction).
// ---------------------------------------------------------------------------
__global__ __launch_bounds__(256) void gather_agg(
    const float* __restrict__ x, const int* __restrict__ offs,
    const int* __restrict__ srcbuf, float* __restrict__ agg_down,
    float* __restrict__ agg_up) {
  const int gid  = blockIdx.x * blockDim.x + threadIdx.x;
  const int id   = gid >> 5;   // 0 .. 2N-1
  const int lane = gid & 31;
  if (id >= NCNT) return;
  const int start = offs[id];
  const int end   = (id == NCNT - 1) ? 2 * NEDGES : offs[id + 1];
  float4 acc = {0.f, 0.f, 0.f, 0.f};
  for (int j = start; j < end; ++j) {
    const int src = srcbuf[j];  // same addr all lanes -> broadcast load
    const float4 v =
        *reinterpret_cast<const float4*>(x + (long)src * HDIM + lane * 4);
    acc.x += v.x;
    acc.y += v.y;
    acc.z += v.z;
    acc.w += v.w;
  }
  float* outp = (id < NNODES) ? (agg_down + (long)id * HDIM)
                              : (agg_up + (long)(id - NNODES) * HDIM);
  *reinterpret_cast<float4*>(outp + lane * 4) = acc;
}

// ---------------------------------------------------------------------------
// Stage B1: tb = ((1+eps)*x + agg) @ w1   [N,128]@[128,32] -> [N,32]
// One wave per 16-row block; BOTH 16-col tiles in registers so the fused
// A panel (x,agg) is read exactly once. 2 independent accumulators.
// ---------------------------------------------------------------------------
__global__ __launch_bounds__(32) void gin_gemm1(
    const float* __restrict__ x, const float* __restrict__ agg,
    const float* __restrict__ eps_p, const float* __restrict__ w1,
    float* __restrict__ tb) {
  const int mtile = blockIdx.x;  // 3125
  const int lane  = threadIdx.x;
  const int l15   = lane & 15;
  const int kp    = lane >> 4;
  const float eps1 = 1.0f + eps_p[0];

  const float* xr = x   + (long)(mtile * 16 + l15) * HDIM;
  const float* ar = agg + (long)(mtile * 16 + l15) * HDIM;

  v8f c0 = {0.f, 0.f, 0.f, 0.f, 0.f, 0.f, 0.f, 0.f};
  v8f c1 = c0;
  for (int kb = 0; kb < HDIM; kb += 4) {
    const int k0 = kb + 2 * kp;
    const float2 xv = *reinterpret_cast<const float2*>(xr + k0);
    const float2 av = *reinterpret_cast<const float2*>(ar + k0);
    v2f a;
    a.x = fmaf(eps1, xv.x, av.x);
    a.y = fmaf(eps1, xv.y, av.y);
    const float* w0 = w1 + k0 * BDIM + l15;
    v2f b0, b1;
    b0.x = w0[0];
    b0.y = w0[BDIM];
    b1.x = w0[16];
    b1.y = w0[BDIM + 16];
    c0 = wmma_f32(a, b0, c0);
    c1 = wmma_f32(a, b1, c1);
  }
#pragma unroll
  for (int r = 0; r < 8; ++r) {
    const int mm = r + 8 * kp;
    float* orow = tb + (long)(mtile * 16 + mm) * BDIM + l15;
    orow[0]  = c0[r];
    orow[16] = c1[r];
  }
}

// ---------------------------------------------------------------------------
// Stage B2: LayerNorm over 32 features + ReLU, in place. One wave per node.
// ---------------------------------------------------------------------------
__global__ __launch_bounds__(256) void ln_relu_32(
    float* __restrict__ tb, const float* __restrict__ g,
    const float* __restrict__ bt) {
  const int gid  = blockIdx.x * blockDim.x + threadIdx.x;
  const int node = gid >> 5;
  const int lane = gid & 31;
  if (node >= NNODES) return;
  const float v = tb[(long)node * BDIM + lane];
  float s = v;
#pragma unroll
  for (int off = 16; off > 0; off >>= 1) s += __shfl_xor(s, off, 32);
  const float mean = s * (1.0f / 32.0f);
  const float dv = v - mean;
  float q = dv * dv;
#pragma unroll
  for (int off = 16; off > 0; off >>= 1) q += __shfl_xor(q, off, 32);
  const float var = q * (1.0f / 32.0f);
  const float o = dv * rsqrtf(var + 1e-5f) * g[lane] + bt[lane];
  tb[(long)node * BDIM + lane] = fmaxf(o, 0.0f);
}

// ---------------------------------------------------------------------------
// Stage B3: h = relu(tb @ w2 + dir)   [N,32]@[32,128] -> [N,128] (pre-LN)
// One wave per 16-row block, all 8 col-tiles in registers (8 independent
// accumulators -> hazard-free back-to-back WMMA issue, A read once).
// ---------------------------------------------------------------------------
__global__ __launch_bounds__(32) void gin_gemm2(
    const float* __restrict__ tb, const float* __restrict__ w2,
    const float* __restrict__ dir, float* __restrict__ hbuf) {
  const int mtile = blockIdx.x;  // 3125
  const int lane  = threadIdx.x;
  const int l15   = lane & 15;
  const int kp    = lane >> 4;

  const float* ar = tb + (long)(mtile * 16 + l15) * BDIM;

  v8f c[8];
#pragma unroll
  for (int nt = 0; nt < 8; ++nt)
    c[nt] = (v8f){0.f, 0.f, 0.f, 0.f, 0.f, 0.f, 0.f, 0.f};

  for (int kb = 0; kb < BDIM; kb += 4) {
    const int k0 = kb + 2 * kp;
    const float2 av = *reinterpret_cast<const float2*>(ar + k0);
    v2f a;
    a.x = av.x;
    a.y = av.y;
    const float* w0 = w2 + k0 * HDIM + l15;
#pragma unroll
    for (int nt = 0; nt < 8; ++nt) {
      v2f b;
      b.x = w0[nt * 16];
      b.y = w0[HDIM + nt * 16];
      c[nt] = wmma_f32(a, b, c[nt]);
    }
  }
#pragma unroll
  for (int nt = 0; nt < 8; ++nt) {
    const int col = nt * 16 + l15;
    const float dvv = dir[col];
#pragma unroll
    for (int r = 0; r < 8; ++r) {
      const int mm = r + 8 * kp;
      hbuf[(long)(mtile * 16 + mm) * HDIM + col] = fmaxf(c[nt][r] + dvv, 0.0f);
    }
  }
}

// ---------------------------------------------------------------------------
// Stage B4: LayerNorm over 128 features, in place. One wave per node.
// ---------------------------------------------------------------------------
__global__ __launch_bounds__(256) void ln_128(
    float* __restrict__ h, const float* __restrict__ g,
    const float* __restrict__ bt) {
  const int gid  = blockIdx.x * blockDim.x + threadIdx.x;
  const int node = gid >> 5;
  const int lane = gid & 31;
  if (node >= NNODES) return;
  float* row = h + (long)node * HDIM;
  const float4 v = *reinterpret_cast<const float4*>(row + lane * 4);
  float s = v.x + v.y + v.z + v.w;
  float q = v.x * v.x + v.y * v.y + v.z * v.z + v.w * v.w;
#pragma unroll
  for (int off = 16; off > 0; off >>= 1) {
    s += __shfl_xor(s, off, 32);
    q += __shfl_xor(q, off, 32);
  }
  const float mean = s * (1.0f / 128.0f);
  const float var = q * (1.0f / 128.0f) - mean * mean;
  const float rs = rsqrtf(var + 1e-5f);
  const float4 gg = *reinterpret_cast<const float4*>(g + lane * 4);
  const float4 bb = *reinterpret_cast<const float4*>(bt + lane * 4);
  float4 o;
  o.x = (v.x - mean) * rs * gg.x + bb.x;
  o.y = (v.y - mean) * rs * gg.y + bb.y;
  o.z = (v.z - mean) * rs * gg.z + bb.z;
  o.w = (v.w - mean) * rs * gg.w + bb.w;
  *reinterpret_cast<float4*>(row + lane * 4) = o;
}

// ---------------------------------------------------------------------------
// Stage C: out = h_down @ cw[0:128,:] + h_up @ cw[128:256,:] + cb
// (concat-free). One wave per 16-row block, 8 col-tiles in registers;
// h_down/h_up panels read exactly once (8x less L2 traffic than 16x16 tiles).
// ---------------------------------------------------------------------------
__global__ __launch_bounds__(32) void combine_gemm(
    const float* __restrict__ hd, const float* __restrict__ hu,
    const float* __restrict__ cw, const float* __restrict__ cb,
    float* __restrict__ out) {
  const int mtile = blockIdx.x;  // 3125
  const int lane  = threadIdx.x;
  const int l15   = lane & 15;
  const int kp    = lane >> 4;

  const float* dr  = hd + (long)(mtile * 16 + l15) * HDIM;
  const float* ur  = hu + (long)(mtile * 16 + l15) * HDIM;
  const float* cwu = cw + (long)HDIM * HDIM;  // rows 128..255

  v8f c[8];
#pragma unroll
  for (int nt = 0; nt < 8; ++nt)
    c[nt] = (v8f){0.f, 0.f, 0.f, 0.f, 0.f, 0.f, 0.f, 0.f};

  for (int kb = 0; kb < HDIM; kb += 4) {
    const int k0 = kb + 2 * kp;
    const float2 advv = *reinterpret_cast<const float2*>(dr + k0);
    const float2 auvv = *reinterpret_cast<const float2*>(ur + k0);
    v2f ad, au;
    ad.x = advv.x;
    ad.y = advv.y;
    au.x = auvv.x;
    au.y = auvv.y;
    const float* wd = cw + k0 * HDIM + l15;
    const float* wu = cwu + k0 * HDIM + l15;
#pragma unroll
    for (int nt = 0; nt < 8; ++nt) {
      v2f bd, bu;
      bd.x = wd[nt * 16];
      bd.y = wd[HDIM + nt * 16];
      bu.x = wu[nt * 16];
      bu.y = wu[HDIM + nt * 16];
      c[nt] = wmma_f32(ad, bd, c[nt]);
      c[nt] = wmma_f32(au, bu, c[nt]);
    }
  }
#pragma unroll
  for (int nt = 0; nt < 8; ++nt) {
    const int col = nt * 16 + l15;
    const float bias = cb[col];
#pragma unroll
    for (int r = 0; r < 8; ++r) {
      const int mm = r + 8 * kp;
      out[(long)(mtile * 16 + mm) * HDIM + col] = c[nt][r] + bias;
    }
  }
}

// ---------------------------------------------------------------------------
extern "C" void kernel_launch(void* const* d_in, const int* in_sizes, int n_in,
                              void* d_out, int out_size, void* d_ws,
                              size_t ws_size, hipStream_t stream) {
  (void)in_sizes; (void)n_in; (void)out_size;

  const float* x      = (const float*)d_in[0];
  const int*   ei     = (const int*)d_in[1];
  const float* eps_dn = (const float*)d_in[2];
  const float* dw1    = (const float*)d_in[3];
  const float* dg1    = (const float*)d_in[4];
  const float* db1    = (const float*)d_in[5];
  const float* dw2    = (const float*)d_in[6];
  const float* eps_up = (const float*)d_in[7];
  const float* uw1    = (const float*)d_in[8];
  const float* ug1    = (const float*)d_in[9];
  const float* ub1    = (const float*)d_in[10];
  const float* uw2    = (const float*)d_in[11];
  const float* ln1_g  = (const float*)d_in[12];
  const float* ln1_b  = (const float*)d_in[13];
  const float* ln2_g  = (const float*)d_in[14];
  const float* ln2_b  = (const float*)d_in[15];
  const float* dir    = (const float*)d_in[16];  // [2,H]
  const float* cw     = (const float*)d_in[17];  // [2H,H]
  const float* cb     = (const float*)d_in[18];
  float*       out    = (float*)d_out;

  // Workspace layout:
  //   floats: agg_down | agg_up | tb | h_down | h_up
  //   ints:   cnt/cursor | offs | srcbuf
  const size_t nh = (size_t)NNODES * HDIM;
  const size_t nbv = (size_t)NNODES * BDIM;
  const size_t nfloats = 4 * nh + nbv;
  const size_t nints = (size_t)2 * NCNT + 2 * NEDGES;
  if (ws_size < nfloats * sizeof(float) + nints * sizeof(int)) return;

  float* agg_down = (float*)d_ws;
  float* agg_up   = agg_down + nh;
  float* tb       = agg_up + nh;
  float* h_down   = tb + nbv;
  float* h_up     = h_down + nh;
  int* cnt    = (int*)(h_up + nh);  // reused as cursor after scan
  int* offs   = cnt + NCNT;
  int* srcbuf = offs + NCNT;

  const int ge = (NEDGES + 255) / 256;

  // --- A: build CSR buckets and gather (no fp atomics) ---
  hipMemsetAsync(cnt, 0, (size_t)NCNT * sizeof(int), stream);
  degree_hist<<<ge, 256, 0, stream>>>(ei, cnt);
  scan_exclusive<<<1, 1024, 0, stream>>>(cnt, offs);
  hipMemcpyAsync(cnt, offs, (size_t)NCNT * sizeof(int),
                 hipMemcpyDeviceToDevice, stream);  // cursor = offs
  fill_buckets<<<ge, 256, 0, stream>>>(ei, cnt, srcbuf);
  gather_agg<<<(NCNT * 32) / 256, 256, 0, stream>>>(x, offs, srcbuf, agg_down,
                                                    agg_up);

  const int gm  = NNODES / 16;          // 3125 row-blocks
  const int gln = (NNODES * 32) / 256;  // 6250

  // --- B: down direction ---
  gin_gemm1<<<gm, 32, 0, stream>>>(x, agg_down, eps_dn, dw1, tb);
  ln_relu_32<<<gln, 256, 0, stream>>>(tb, dg1, db1);
  gin_gemm2<<<gm, 32, 0, stream>>>(tb, dw2, dir, h_down);
  ln_128<<<gln, 256, 0, stream>>>(h_down, ln1_g, ln1_b);

  // --- B: up direction (reversed edges live in the second bucket half) ---
  gin_gemm1<<<gm, 32, 0, stream>>>(x, agg_up, eps_up, uw1, tb);
  ln_relu_32<<<gln, 256, 0, stream>>>(tb, ug1, ub1);
  gin_gemm2<<<gm, 32, 0, stream>>>(tb, uw2, dir + HDIM, h_up);
  ln_128<<<gln, 256, 0, stream>>>(h_up, ln2_g, ln2_b);

  // --- C: final combine ---
  combine_gemm<<<gm, 32, 0, stream>>>(h_down, h_up, cw, cb, out);
}